// My_L2loss_58720792871303
// MI455X (gfx1250) — compile-verified
//
#include <hip/hip_runtime.h>

// ---------------------------------------------------------------------------
// loss = sum_{i,j} (x[i,j] - h)^2 * h,  h = expf(-sqrt((i-l0)^2+(j-l1)^2))
//
// Key optimization: h == 0.0f *exactly* (fp32 underflow, incl. denormal range)
// whenever dist > ~104, and every term is multiplied by h, so pixels outside a
// 104-radius disk around `label` contribute exactly zero and need not be read.
// We process a clamped 257x260 window (half-width 128, cols 16B-aligned):
// ~267 KB of traffic instead of 64 MB -> latency-bound instead of HBM-bound.
//
// CDNA5: wave32 reduction done with one V_WMMA_F32_16X16X4_F32 (B = ones =>
// D rows are row-sums of A) + one shfl_xor(16), instead of a 5-step shuffle
// tree. Deterministic two-phase block reduction (no float atomics).
// ---------------------------------------------------------------------------

typedef float v2f __attribute__((ext_vector_type(2)));
typedef float v8f __attribute__((ext_vector_type(8)));

namespace {
constexpr int kH = 4096, kW = 4096;
constexpr int kHalf    = 128;               // window half-width (>= 104 + align slack)
constexpr int kRows    = 2 * kHalf + 1;     // 257 rows
constexpr int kColQ    = 65;                // 65 float4 quads = 260 columns
constexpr int kTotQ    = kRows * kColQ;     // 16705 quads
constexpr int kThreads = 256;               // 8 wave32 per block
constexpr int kBlocks  = (kTotQ + kThreads - 1) / kThreads;  // 66
}

// Full 32-lane sum via one f32 WMMA. A[m][k0] = acc(lane), other K slot = 0,
// B = all ones  =>  D[m][n] = acc(lane m) + acc(lane m+16)  (independent of n).
// Per-lane sum of the 8 D VGPRs gives the half-row-group sums; one xor-16
// shuffle combines them. Requires EXEC all-ones (callers guarantee full waves,
// converged control flow).
__device__ __forceinline__ float wave_reduce_wmma(float acc) {
  v2f a; a[0] = acc;  a[1] = 0.0f;
  v2f b; b[0] = 1.0f; b[1] = 1.0f;
  v8f c = {};
  c = __builtin_amdgcn_wmma_f32_16x16x4_f32(
      /*neg_a=*/false, a, /*neg_b=*/false, b,
      /*c_mod=*/(short)0, c, /*reuse_a=*/false, /*reuse_b=*/false);
  float s = ((c[0] + c[1]) + (c[2] + c[3])) + ((c[4] + c[5]) + (c[6] + c[7]));
  s += __shfl_xor(s, 16);   // lanes 0-15 hold M=0..7 sums, 16-31 hold M=8..15
  return s;                 // every lane: full wave sum
}

__global__ __launch_bounds__(kThreads)
void window_loss_partial(const float* __restrict__ x,
                         const int*   __restrict__ label,
                         float*       __restrict__ partial) {
  const int l0 = label[0];
  const int l1 = label[1];

  // Clamp window into the tensor; always full-size so coverage of the
  // nonzero-h disk is preserved at the borders. Columns aligned to 4 floats.
  int i0 = l0 - kHalf;
  i0 = (i0 < 0) ? 0 : i0;
  i0 = (i0 > kH - kRows) ? (kH - kRows) : i0;
  int j0 = l1 - kHalf;
  j0 = (j0 < 0) ? 0 : j0;
  j0 = (j0 > kW - 4 * kColQ) ? (kW - 4 * kColQ) : j0;
  j0 &= ~3;

  float acc = 0.0f;
  const int stride = gridDim.x * blockDim.x;
  for (int q = blockIdx.x * blockDim.x + threadIdx.x; q < kTotQ; q += stride) {
    const int r  = q / kColQ;
    const int cq = q - r * kColQ;
    const int i  = i0 + r;
    const int j  = j0 + 4 * cq;

    const float4 xv = *reinterpret_cast<const float4*>(x + (size_t)i * kW + j);
    const float di  = (float)(i - l0);
    const float di2 = di * di;
    const float xs[4] = {xv.x, xv.y, xv.z, xv.w};
#pragma unroll
    for (int t = 0; t < 4; ++t) {
      const float dj   = (float)(j + t - l1);
      const float dist = sqrtf(di2 + dj * dj);
      const float hv   = expf(-dist);          // exactly 0 beyond the disk
      const float d    = xs[t] - hv;
      acc = fmaf(d * d, hv, acc);
    }
  }

  // Converged here; full 256-thread blocks => EXEC all-ones for WMMA.
  const float wsumv = wave_reduce_wmma(acc);

  __shared__ float wsum[kThreads / 32];
  const int lane = threadIdx.x & 31;
  const int wid  = threadIdx.x >> 5;
  if (lane == 0) wsum[wid] = wsumv;
  __syncthreads();
  if (threadIdx.x == 0) {
    float t = 0.0f;
#pragma unroll
    for (int w = 0; w < kThreads / 32; ++w) t += wsum[w];
    partial[blockIdx.x] = t;   // fixed slot -> deterministic
  }
}

// One wave, fixed-order accumulation of block partials -> bitwise-stable
// result across replays. Reuses the WMMA wave reducer.
__global__ __launch_bounds__(32)
void reduce_partials(const float* __restrict__ partial, int n,
                     float* __restrict__ out) {
  float t = 0.0f;
  for (int idx = threadIdx.x; idx < n; idx += 32) t += partial[idx];
  const float s = wave_reduce_wmma(t);
  if (threadIdx.x == 0) out[0] = s;
}

extern "C" void kernel_launch(void* const* d_in, const int* in_sizes, int n_in,
                              void* d_out, int out_size, void* d_ws, size_t ws_size,
                              hipStream_t stream) {
  (void)in_sizes; (void)n_in; (void)out_size; (void)ws_size;
  const float* x     = (const float*)d_in[0];
  const int*   label = (const int*)d_in[1];   // harness: integer -> const int*
  float*       out   = (float*)d_out;
  float*       ws    = (float*)d_ws;          // kBlocks floats of scratch

  window_loss_partial<<<kBlocks, kThreads, 0, stream>>>(x, label, ws);
  reduce_partials<<<1, 32, 0, stream>>>(ws, kBlocks, out);
}